// GNN_23630910062676
// MI455X (gfx1250) — compile-verified
//
#include <hip/hip_runtime.h>

typedef float v2f __attribute__((ext_vector_type(2)));
typedef float v8f __attribute__((ext_vector_type(8)));

// ---------------------------------------------------------------------------
// Degree / normalization kernels
// ---------------------------------------------------------------------------
__global__ void gcn_init_deg(float* __restrict__ deg, int n) {
    int i = blockIdx.x * blockDim.x + threadIdx.x;
    if (i < n) deg[i] = 1.0f;   // self-loop contributes 1 to every node's degree
}

__global__ void gcn_count_deg(const int* __restrict__ dst, float* __restrict__ deg, int e) {
    int i = blockIdx.x * blockDim.x + threadIdx.x;
    if (i < e) atomicAdd(&deg[dst[i]], 1.0f);
}

__global__ void gcn_rsqrt(const float* __restrict__ deg, float* __restrict__ dinv, int n) {
    int i = blockIdx.x * blockDim.x + threadIdx.x;
    if (i < n) {
        float d = deg[i];
        dinv[i] = d > 0.0f ? rsqrtf(d) : 0.0f;
    }
}

// ---------------------------------------------------------------------------
// WMMA f32 GEMM: C[M x Ncols] = op(A[M x K]) @ B[K x Ncols]
// One wave32 computes one 16x16 tile of C using V_WMMA_F32_16X16X4_F32.
// RELU applies max(0,.) to A elements on load (fuses layer-1 activation).
// Requires M % 16 == 0, K % 4 == 0, Ncols % 16 == 0.
// ---------------------------------------------------------------------------
template <bool RELU>
__global__ void gcn_gemm_wmma(const float* __restrict__ A,
                              const float* __restrict__ B,
                              float* __restrict__ C,
                              int M, int K, int Ncols) {
    const int wave = (blockIdx.x * blockDim.x + threadIdx.x) >> 5;
    const int lane = threadIdx.x & 31;
    const int tilesN = Ncols >> 4;
    const int rowTile = wave / tilesN;
    const int colTile = wave - rowTile * tilesN;
    if (rowTile * 16 >= M) return;               // wave-uniform: EXEC stays all-1s

    const int rowBase = rowTile << 4;
    const int colBase = colTile << 4;
    const int m     = lane & 15;                 // row (A) / col (B,C) within tile
    const int khalf = (lane >> 4) << 1;          // lanes 0-15 -> K+0,1 ; lanes 16-31 -> K+2,3

    v8f acc = {};
    const float* Arow = A + (size_t)(rowBase + m) * K;
    const float* Bcol = B + colBase + m;

    for (int k = 0; k < K; k += 4) {
        float a0 = Arow[k + khalf];
        float a1 = Arow[k + khalf + 1];
        if (RELU) { a0 = fmaxf(a0, 0.0f); a1 = fmaxf(a1, 0.0f); }
        v2f a; a.x = a0; a.y = a1;

        v2f b;
        b.x = Bcol[(size_t)(k + khalf)     * Ncols];
        b.y = Bcol[(size_t)(k + khalf + 1) * Ncols];

        // 8 args: (neg_a, A, neg_b, B, c_mod, C, reuse_a, reuse_b)
        acc = __builtin_amdgcn_wmma_f32_16x16x4_f32(
            false, a, false, b, (short)0, acc, false, false);
    }

    // C/D layout: VGPR r -> M = r (lanes 0-15) / r+8 (lanes 16-31), N = lane&15
    const int rOff = (lane >> 4) << 3;
    float* Crow = C + (size_t)(rowBase + rOff) * Ncols + colBase + m;
#pragma unroll
    for (int r = 0; r < 8; ++r)
        Crow[(size_t)r * Ncols] = acc[r];
}

// ---------------------------------------------------------------------------
// out[n][f] = bias[f] + h[n][f] * dinv[n]^2      (bias + self-loop message)
// ---------------------------------------------------------------------------
template <int F>
__global__ void gcn_init_out(const float* __restrict__ h,
                             const float* __restrict__ dinv,
                             const float* __restrict__ bias,
                             float* __restrict__ out, int n) {
    int id = blockIdx.x * blockDim.x + threadIdx.x;
    int node = id / F;
    int f    = id - node * F;
    if (node < n) {
        float di = dinv[node];
        out[id] = bias[f] + h[id] * (di * di);
    }
}

// ---------------------------------------------------------------------------
// Edge scatter: out[dst] += h[src] * dinv[src]*dinv[dst]
// One thread per (edge, float4 chunk): coalesced F*4-byte gather per edge,
// f32 atomics to contiguous addresses (L2-resident).
// ---------------------------------------------------------------------------
template <int F>
__global__ void gcn_edge_scatter(const int* __restrict__ src,
                                 const int* __restrict__ dst,
                                 const float* __restrict__ dinv,
                                 const float* __restrict__ h,
                                 float* __restrict__ out, int e) {
    constexpr int CPE = F / 4;                   // float4 chunks per edge
    int id = blockIdx.x * blockDim.x + threadIdx.x;
    int edge = id / CPE;
    int c    = id - edge * CPE;
    if (edge >= e) return;
    int s = src[edge];
    int d = dst[edge];
    float w = dinv[s] * dinv[d];
    const float4 v = *((const float4*)(h + (size_t)s * F) + c);
    float* o = out + (size_t)d * F + (c << 2);
    atomicAdd(o + 0, v.x * w);
    atomicAdd(o + 1, v.y * w);
    atomicAdd(o + 2, v.z * w);
    atomicAdd(o + 3, v.w * w);
}

// ---------------------------------------------------------------------------
// Launch
// ---------------------------------------------------------------------------
extern "C" void kernel_launch(void* const* d_in, const int* in_sizes, int n_in,
                              void* d_out, int out_size, void* d_ws, size_t ws_size,
                              hipStream_t stream) {
    constexpr int D_IN = 64, D_HID = 128, D_OUT = 64;

    const float* x  = (const float*)d_in[0];
    const int*   ei = (const int*)  d_in[1];
    const float* W1 = (const float*)d_in[2];
    const float* b1 = (const float*)d_in[3];
    const float* W2 = (const float*)d_in[4];
    const float* b2 = (const float*)d_in[5];
    float* out = (float*)d_out;

    const int N = in_sizes[0] / D_IN;
    const int E = in_sizes[1] / 2;
    const int* src = ei;       // edge_index[0]
    const int* dst = ei + E;   // edge_index[1]

    // Workspace layout (floats): deg[N] | dinv[N] | h1[N*128] | out1[N*128]
    // h1's buffer is reused for h2[N*64] after the layer-1 scatter consumes h1.
    float* ws   = (float*)d_ws;
    float* deg  = ws;
    float* dinv = deg + N;
    float* h1   = dinv + N;                  // N*D_HID (later reused as h2, N*D_OUT)
    float* out1 = h1 + (size_t)N * D_HID;    // N*D_HID

    auto cdiv = [](long a, long b) { return (int)((a + b - 1) / b); };
    const int BLK = 256;

    // --- normalization ---
    gcn_init_deg <<<cdiv(N, BLK), BLK, 0, stream>>>(deg, N);
    gcn_count_deg<<<cdiv(E, BLK), BLK, 0, stream>>>(dst, deg, E);
    gcn_rsqrt    <<<cdiv(N, BLK), BLK, 0, stream>>>(deg, dinv, N);

    // --- layer 1: h1 = x @ W1 ---
    {
        long waves = (long)(N / 16) * (D_HID / 16);
        gcn_gemm_wmma<false><<<cdiv(waves * 32, BLK), BLK, 0, stream>>>(
            x, W1, h1, N, D_IN, D_HID);
    }
    gcn_init_out<D_HID><<<cdiv((long)N * D_HID, BLK), BLK, 0, stream>>>(h1, dinv, b1, out1, N);
    gcn_edge_scatter<D_HID><<<cdiv((long)E * (D_HID / 4), BLK), BLK, 0, stream>>>(
        src, dst, dinv, h1, out1, E);

    // --- layer 2: h2 = relu(out1) @ W2  (ReLU fused into A load; h2 -> h1's buffer) ---
    {
        long waves = (long)(N / 16) * (D_OUT / 16);
        gcn_gemm_wmma<true><<<cdiv(waves * 32, BLK), BLK, 0, stream>>>(
            out1, W2, h1, N, D_HID, D_OUT);
    }
    gcn_init_out<D_OUT><<<cdiv((long)N * D_OUT, BLK), BLK, 0, stream>>>(h1, dinv, b2, out, N);
    gcn_edge_scatter<D_OUT><<<cdiv((long)E * (D_OUT / 4), BLK), BLK, 0, stream>>>(
        src, dst, dinv, h1, out, E);
}